// ModelNew_3556232922055
// MI455X (gfx1250) — compile-verified
//
#include <hip/hip_runtime.h>
#include <hip/hip_bf16.h>

typedef __attribute__((ext_vector_type(2))) float v2f;
typedef __attribute__((ext_vector_type(8))) float v8f;

// Problem constants (from reference): B=8, Cin=8, H=W=128, Cout=32, K=3
#define OH 126
#define OW 126
#define PIX (OH * OW)          // 15876 pixels per batch
#define NPIX_TOT (8 * PIX)     // 127008
#define TILES_PER_B 993        // ceil(15876 / 16)
#define KTOT 72                // Cin*3*3

// im2col element k = ci*9 + kh*3 + kw  ->  byte-free offset into x (per batch, per pixel base)
static __device__ __forceinline__ int xoff(int k) {
    int ci = k / 9;
    int r  = k - ci * 9;
    int kh = r / 3;
    int kw = r - kh * 3;
    return ci * (128 * 128) + kh * 128 + kw;
}

// ---------------------------------------------------------------------------
// Kernel 1: VALID 3x3 conv for the first 8 output channels, via
// V_WMMA_F32_16X16X4_F32 (exact f32). One wave = one 16-pixel x 16-col tile
// (cols 0..7 = real out-channels, 8..15 = zero weights).
// Output layout: ysub[b][pixel][c], c contiguous (8 floats = 32B per pixel).
// ---------------------------------------------------------------------------
__global__ __launch_bounds__(256) void conv_wmma_f32(
        const float* __restrict__ x,       // [8][8][128][128]
        const float* __restrict__ weight,  // [32][8][3][3]
        const float* __restrict__ bias,    // [32]
        float* __restrict__ ysub)          // [8][15876][8]
{
    const int lane   = threadIdx.x & 31;
    const int wave   = threadIdx.x >> 5;
    const int tile   = blockIdx.x * 8 + wave;        // 0 .. 7943 (993 blocks * 8 waves)
    const int b      = tile / TILES_PER_B;
    const int tileIB = tile - b * TILES_PER_B;

    const int mLocal = lane & 15;                    // M (pixel-in-tile) / N (out-channel)
    const int kHalf  = lane >> 4;                    // 0: K 0..1, 1: K 2..3 (per 32-bit A/B layout)

    // Pixel handled by this lane as an A-matrix row (clamped for the partial last tile;
    // WMMA needs EXEC all-1s, so we compute redundantly and mask only the store).
    int p  = tileIB * 16 + mLocal;
    int pc = p < PIX ? p : (PIX - 1);
    const int h = pc / OW;
    const int w = pc - h * OW;
    const float* xb = x + b * (8 * 128 * 128) + h * 128 + w;

    // B-matrix column n = mLocal: weight row (out-channel) if n < 8, else zero.
    const float* wn   = weight + mLocal * KTOT;      // in-bounds for mLocal<16 (32 rows exist)
    const bool   colOn = (mLocal < 8);

    v8f acc = {};
    #pragma unroll
    for (int kk = 0; kk < 18; ++kk) {                // 18 * K4 = 72
        const int kEven = kk * 4 + (kHalf ? 2 : 0);  // this lane's first K index
        v2f a, bm;
        a.x = xb[kHalf ? xoff(kk * 4 + 2) : xoff(kk * 4 + 0)];
        a.y = xb[kHalf ? xoff(kk * 4 + 3) : xoff(kk * 4 + 1)];
        bm.x = colOn ? wn[kEven]     : 0.0f;
        bm.y = colOn ? wn[kEven + 1] : 0.0f;
        // (neg_a, A, neg_b, B, c_mod, C, reuse_a, reuse_b)
        acc = __builtin_amdgcn_wmma_f32_16x16x4_f32(
                false, a, false, bm, (short)0, acc, false, false);
    }

    // D layout: VGPR r -> (M = r + 8*kHalf, N = mLocal). Store only real channels.
    #pragma unroll
    for (int r = 0; r < 8; ++r) {
        const int pm = tileIB * 16 + r + kHalf * 8;
        if (pm < PIX && mLocal < 8) {
            ysub[((size_t)b * PIX + pm) * 8 + mLocal] = acc[r] + bias[mLocal];
        }
    }
}

// ---------------------------------------------------------------------------
// Kernel 2: out[b,h,w] = 2 * min_o max_{valid kh,kw,c} ysub[b,c,h+kh,w+kw]*W[o,c,kh,kw]
// One thread per output pixel. Interior path fully unrolled (taps in VGPRs,
// weights are lane-uniform -> scalar loads). Border path uses clamped bounds.
// ---------------------------------------------------------------------------
__global__ __launch_bounds__(256) void maxmul_min_kernel(
        const float* __restrict__ ysub,    // [8][15876][8]
        const float* __restrict__ weight,  // [32][8][3][3]
        float* __restrict__ out)           // [8][1][126][126]
{
    const int idx = blockIdx.x * 256 + threadIdx.x;
    if (idx >= NPIX_TOT) return;

    const int b = idx / PIX;
    const int p = idx - b * PIX;
    const int h = p / OW;
    const int w = p - h * OW;
    const float* yb = ysub + (size_t)b * PIX * 8;

    const float NEG = -__builtin_inff();
    float mn;

    if (h < OH - 2 && w < OW - 2) {
        // Interior: all 9 taps valid.
        float t[9][8];
        #pragma unroll
        for (int kh = 0; kh < 3; ++kh) {
            #pragma unroll
            for (int kw = 0; kw < 3; ++kw) {
                const float4* tp =
                    (const float4*)(yb + ((size_t)((h + kh) * OW + (w + kw))) * 8);
                float4 lo = tp[0], hi = tp[1];
                float* d = t[kh * 3 + kw];
                d[0] = lo.x; d[1] = lo.y; d[2] = lo.z; d[3] = lo.w;
                d[4] = hi.x; d[5] = hi.y; d[6] = hi.z; d[7] = hi.w;
            }
        }
        mn = __builtin_inff();
        #pragma unroll 4
        for (int o = 0; o < 32; ++o) {
            const float* wr = weight + o * KTOT;   // 72 contiguous, lane-uniform
            float a = NEG;
            #pragma unroll
            for (int kh = 0; kh < 3; ++kh)
                #pragma unroll
                for (int kw = 0; kw < 3; ++kw)
                    #pragma unroll
                    for (int c = 0; c < 8; ++c)
                        a = fmaxf(a, t[kh * 3 + kw][c] * wr[c * 9 + kh * 3 + kw]);
            mn = fminf(mn, a);
        }
    } else {
        // Border: only taps with h+kh<OH, w+kw<OW participate (-inf padding in ref).
        const int nkh = min(3, OH - h);
        const int nkw = min(3, OW - w);
        float t[9][8];
        for (int kh = 0; kh < nkh; ++kh)
            for (int kw = 0; kw < nkw; ++kw) {
                const float4* tp =
                    (const float4*)(yb + ((size_t)((h + kh) * OW + (w + kw))) * 8);
                float4 lo = tp[0], hi = tp[1];
                float* d = t[kh * 3 + kw];
                d[0] = lo.x; d[1] = lo.y; d[2] = lo.z; d[3] = lo.w;
                d[4] = hi.x; d[5] = hi.y; d[6] = hi.z; d[7] = hi.w;
            }
        mn = __builtin_inff();
        for (int o = 0; o < 32; ++o) {
            const float* wr = weight + o * KTOT;
            float a = NEG;
            for (int kh = 0; kh < nkh; ++kh)
                for (int kw = 0; kw < nkw; ++kw)
                    #pragma unroll
                    for (int c = 0; c < 8; ++c)
                        a = fmaxf(a, t[kh * 3 + kw][c] * wr[c * 9 + kh * 3 + kw]);
            mn = fminf(mn, a);
        }
    }

    out[idx] = 2.0f * mn;   // SCALE_FACTOR
}

extern "C" void kernel_launch(void* const* d_in, const int* in_sizes, int n_in,
                              void* d_out, int out_size, void* d_ws, size_t ws_size,
                              hipStream_t stream) {
    const float* x      = (const float*)d_in[0];   // (8,8,128,128)
    const float* weight = (const float*)d_in[1];   // (32,8,3,3)
    const float* bias   = (const float*)d_in[2];   // (32,)
    float* out  = (float*)d_out;                   // (8,1,126,126) = 127008 floats
    float* ysub = (float*)d_ws;                    // needs 8*15876*8*4 B ~= 4.1 MB

    // Phase 1: conv (first 8 out-channels) via f32 WMMA. 993 blocks * 8 waves = 7944 tiles.
    conv_wmma_f32<<<dim3(TILES_PER_B), dim3(256), 0, stream>>>(x, weight, bias, ysub);

    // Phase 2: max-mul / min reduction, one thread per output pixel.
    maxmul_min_kernel<<<dim3((NPIX_TOT + 255) / 256), dim3(256), 0, stream>>>(ysub, weight, out);
}